// PatchCCAConv_12661563589195
// MI455X (gfx1250) — compile-verified
//
#include <hip/hip_runtime.h>
#include <hip/hip_bf16.h>

typedef float v2f __attribute__((ext_vector_type(2)));
typedef float v8f __attribute__((ext_vector_type(8)));

#define HSIZE 512
#define WSIZE 512
#define CCH   6      // channels per input
#define NCH   12     // x channels + y channels
#define EPSV  1e-6f

__device__ __forceinline__ int reflect_idx(int g, int n) {
    if (g < 0) g = -g;
    if (g >= n) g = 2 * n - 2 - g;
    return g;
}

// packed lower-triangular index, j <= i
__device__ __forceinline__ constexpr int lt(int i, int j) { return i * (i + 1) / 2 + j; }

// Cholesky of 6x6 SPD (packed lower, 21 entries) then invert the lower factor.
__device__ __forceinline__ void chol_inv6(const float* A, float* M) {
    float L[21];
#pragma unroll
    for (int c = 0; c < 6; ++c) {
        float d = A[lt(c, c)];
#pragma unroll
        for (int j = 0; j < c; ++j) d -= L[lt(c, j)] * L[lt(c, j)];
        float s = sqrtf(d);
        L[lt(c, c)] = s;
        float inv = 1.0f / s;
#pragma unroll
        for (int rr = c + 1; rr < 6; ++rr) {
            float v = A[lt(rr, c)];
#pragma unroll
            for (int j = 0; j < c; ++j) v -= L[lt(rr, j)] * L[lt(c, j)];
            L[lt(rr, c)] = v * inv;
        }
    }
    // invert lower-triangular L -> M
#pragma unroll
    for (int i = 0; i < 6; ++i) {
        float di = 1.0f / L[lt(i, i)];
        M[lt(i, i)] = di;
#pragma unroll
        for (int j = 0; j < i; ++j) {
            float sum = 0.0f;
#pragma unroll
            for (int p = j; p < i; ++p) sum += L[lt(i, p)] * M[lt(p, j)];
            M[lt(i, j)] = -sum * di;
        }
    }
}

// Block = 64 threads (2 waves). Tile = 8x8 output pixels.
// Phase 0: cooperative reflect-padded patch load into LDS.
// Phase 1: per-wave, per-pixel augmented Gram (13x13 inside 16x16) via chained
//          V_WMMA_F32_16X16X4_F32: G = A * A^T where A rows = [x0..x5,y0..y5,1,0,0,0]
//          over the k*k window cells. Row 12 of G yields the window sums (means).
// Phase 2: thread-per-pixel covariance -> Cholesky -> tri-inverse -> diag(triple product).
__global__ __launch_bounds__(64) void patch_cca_kernel(const float* __restrict__ xin,
                                                       const float* __restrict__ yin,
                                                       float* __restrict__ out,
                                                       int r, int rIdx) {
    __shared__ float patch[NCH][18][17];   // [channel][row][col], margin rows for pad steps
    __shared__ float gram[64][91];         // packed symmetric 13x13 per pixel

    const int k   = 2 * r + 1;
    const int k2  = k * k;
    const int pw  = 8 + 2 * r;             // patch width actually used
    const int tid = threadIdx.x;
    const int b   = blockIdx.z;
    const int x0  = blockIdx.x * 8;
    const int y0  = blockIdx.y * 8;

    // ---- phase 0: load patch (reflect padding) ----
    const int tot = NCH * pw * pw;
    for (int idx = tid; idx < tot; idx += 64) {
        int ch  = idx / (pw * pw);
        int rem = idx - ch * pw * pw;
        int py  = rem / pw;
        int px  = rem - py * pw;
        int gx  = reflect_idx(x0 - r + px, WSIZE);
        int gy  = reflect_idx(y0 - r + py, HSIZE);
        const float* src = (ch < CCH) ? xin : yin;
        int c = (ch < CCH) ? ch : ch - CCH;
        patch[ch][py][px] = src[((size_t)(b * CCH + c) * HSIZE + gy) * WSIZE + gx];
    }
    __syncthreads();

    // ---- phase 1: WMMA Grams (each wave owns 32 pixels) ----
    const int wave = tid >> 5;
    const int lane = tid & 31;
    const int ch   = lane & 15;            // A-operand row (M) and B-operand col (N)
    const int half = lane >> 4;
    const int chc  = (ch < NCH) ? ch : (NCH - 1);   // safe LDS channel index
    const int nsteps = (k2 + 3) >> 2;

    for (int p = wave * 32; p < wave * 32 + 32; ++p) {
        const int py = p >> 3;
        const int px = p & 7;
        v8f acc = {0.f, 0.f, 0.f, 0.f, 0.f, 0.f, 0.f, 0.f};

        int cA = 2 * half;                 // linear window-cell index of this lane's 1st value
        int wy = 0, wx = 2 * half;         // its (row, col) inside the k x k window
        for (int s = 0; s < nsteps; ++s) {
            int wxb = wx + 1, wyb = wy;
            if (wxb >= k) { wxb -= k; wyb += 1; }
            // branchless: always load a valid LDS address, then select
            float pa = patch[chc][py + wy ][px + wx ];
            float pb = patch[chc][py + wyb][px + wxb];
            float va = (cA     < k2) ? ((ch < NCH) ? pa : (ch == NCH ? 1.0f : 0.0f)) : 0.0f;
            float vb = (cA + 1 < k2) ? ((ch < NCH) ? pb : (ch == NCH ? 1.0f : 0.0f)) : 0.0f;
            v2f a; a.x = va; a.y = vb;
            // For 16x16x4 f32, A (M in lanes, K in vgpr/half) and B (N in lanes, K in
            // vgpr/half) have identical per-lane data when B = A^T -> pass `a` twice.
            acc = __builtin_amdgcn_wmma_f32_16x16x4_f32(
                false, a, false, a, (short)0, acc, false, false);
            cA += 4;
            wx += 4;
            if (wx >= k) { wx -= k; wy += 1; }
        }
        // scatter unique entries m <= n <= 12 (packed idx = n(n+1)/2 + m)
#pragma unroll
        for (int v = 0; v < 8; ++v) {
            int m = v + 8 * half;
            int n = ch;
            if (m <= n && n <= 12) gram[p][n * (n + 1) / 2 + m] = acc[v];
        }
    }
    __syncthreads();

    // ---- phase 2: per-pixel 6x6 linear algebra ----
    const float invK2 = 1.0f / (float)k2;
    const float* g = gram[tid];

    float mean[12];
#pragma unroll
    for (int j = 0; j < 12; ++j) mean[j] = g[78 + j] * invK2;  // (m=j, n=12)

    float cxx[21], cyy[21];
#pragma unroll
    for (int i = 0; i < 6; ++i)
#pragma unroll
        for (int j = 0; j <= i; ++j) {
            float e = (i == j) ? EPSV : 0.0f;
            cxx[lt(i, j)] = g[lt(i, j)] * invK2 - mean[i] * mean[j] + e;
            cyy[lt(i, j)] = g[lt(i + 6, j + 6)] * invK2 - mean[i + 6] * mean[j + 6] + e;
        }
    float cxy[6][6];
#pragma unroll
    for (int i = 0; i < 6; ++i)
#pragma unroll
        for (int j = 0; j < 6; ++j)
            cxy[i][j] = g[(j + 6) * (j + 7) / 2 + i] * invK2 - mean[i] * mean[j + 6];

    float Lxi[21], Lyi[21];
    chol_inv6(cxx, Lxi);
    chol_inv6(cyy, Lyi);

    // diag(Lxi * Cxy * Lyi); Lxi lower (j<=i), Lyi lower (k>=i)
    float sim = 0.0f;
#pragma unroll
    for (int i = 0; i < 6; ++i) {
        float di = 0.0f;
#pragma unroll
        for (int j = 0; j <= i; ++j) {
            float t = 0.0f;
#pragma unroll
            for (int kk = i; kk < 6; ++kk) t += cxy[j][kk] * Lyi[lt(kk, i)];
            di += Lxi[lt(i, j)] * t;
        }
        sim += fabsf(di);
    }
    sim *= (1.0f / 6.0f);

    const int gy = y0 + (tid >> 3);
    const int gx = x0 + (tid & 7);
    out[(((size_t)b * HSIZE + gy) * WSIZE + gx) * 2 + rIdx] = sim;
}

extern "C" void kernel_launch(void* const* d_in, const int* in_sizes, int n_in,
                              void* d_out, int out_size, void* d_ws, size_t ws_size,
                              hipStream_t stream) {
    (void)in_sizes; (void)n_in; (void)out_size; (void)d_ws; (void)ws_size;
    const float* x = (const float*)d_in[0];
    const float* y = (const float*)d_in[1];
    float* out = (float*)d_out;

    dim3 grid(WSIZE / 8, HSIZE / 8, 2);
    dim3 block(64);
    patch_cca_kernel<<<grid, block, 0, stream>>>(x, y, out, 2, 0);  // RADII[0]=2
    patch_cca_kernel<<<grid, block, 0, stream>>>(x, y, out, 4, 1);  // RADII[1]=4
}